// MeanAggregator_30099130811057
// MI455X (gfx1250) — compile-verified
//
#include <hip/hip_runtime.h>
#include <stdint.h>

// ---------------------------------------------------------------------------
// MeanAggregator for MI455X (gfx1250), wave32.
//
// One wave32 per node b:
//   lane k: w_k = (t_k <= 100) ? exp((t_k-100)/100) : 0
//   tot    = wave-reduce(w);  tot = (tot==0) ? 1 : tot;  wn_k = w_k / tot
//   denom  = 1 + sum_k wn_k   (accumulated scalar-side from readlanes)
//   out[b,:] = (F[nodes[b],:] + sum_k wn_k * F[neigh[b,k],:]) / denom
//
// Memory-bound (~870 MB of gathers @ 23.3 TB/s, ~15-37 us depending on L2
// hit rate on the 256 MB feature table vs 192 MB L2 with ~3.2x row reuse).
// Each neighbor row is a contiguous 512B line == one wave-wide
// GLOBAL_LOAD_ASYNC_TO_LDS_B128 (lane i moves 16B). All 32 asyncs issue
// first (ASYNCcnt cap 63), weight math runs in their shadow, completion is
// consumed with staged s_wait_asynccnt (async loads retire in order), then
// ds_load_b128 + v_fmac with the weight broadcast via v_readlane (SGPR
// operand). Streamed-once data (ids/times loads, output store) is marked
// non-temporal to preserve L2 for the reuse-bearing feature table.
// ---------------------------------------------------------------------------

#ifndef __has_builtin
#define __has_builtin(x) 0
#endif

namespace {
constexpr int   K_NEIGH         = 32;    // neighbors per node == wave32 lanes
constexpr int   D_FEAT          = 128;   // feature dim = 32 lanes * 4 floats
constexpr int   WAVES_PER_BLOCK = 4;     // 4 waves * 16KB staging = 64KB LDS
constexpr int   BLOCK           = WAVES_PER_BLOCK * 32;
constexpr float T_NOW           = 100.0f;
constexpr float INV_TAU         = 0.01f;
} // namespace

// Native clang vectors (HIP's float4 is a struct; builtins need real vectors).
typedef float v4f __attribute__((ext_vector_type(4)));
typedef int   v4i __attribute__((vector_size(16)));
typedef __attribute__((address_space(1))) v4i* g1_v4i_p;
typedef __attribute__((address_space(3))) v4i* l3_v4i_p;

// One wave-wide async copy: 32 lanes x 16B = one 512B feature row -> LDS.
__device__ __forceinline__ void async_gather_b128(const float* gsrc, float* ldst) {
#if __has_builtin(__builtin_amdgcn_global_load_async_to_lds_b128)
  g1_v4i_p g = reinterpret_cast<g1_v4i_p>(reinterpret_cast<uintptr_t>(gsrc));
  l3_v4i_p l = reinterpret_cast<l3_v4i_p>((uint32_t)reinterpret_cast<uintptr_t>(ldst));
  __builtin_amdgcn_global_load_async_to_lds_b128(g, l, /*offset=*/0, /*cpol=*/0);
#else
  asm volatile("global_load_async_to_lds_b128 %0, %1, off"
               :
               : "v"((uint32_t)reinterpret_cast<uintptr_t>(ldst)),
                 "v"(reinterpret_cast<uintptr_t>(gsrc))
               : "memory");
#endif
}

template <int N>
__device__ __forceinline__ void wait_asynccnt() {
#if __has_builtin(__builtin_amdgcn_s_wait_asynccnt)
  __builtin_amdgcn_s_wait_asynccnt(N);
#else
  asm volatile("s_wait_asynccnt %0" : : "i"(N) : "memory");
#endif
  asm volatile("" ::: "memory"); // keep LDS reads below the wait
}

// Uniform (SGPR) broadcast of lane `k` — v_readlane_b32, no LDS traffic.
__device__ __forceinline__ int broadcast_i32(int v, int k) {
  return __builtin_amdgcn_readlane(v, k);
}
__device__ __forceinline__ float broadcast_f32(float v, int k) {
  return __uint_as_float(__builtin_amdgcn_readlane(__float_as_uint(v), k));
}

__global__ void __launch_bounds__(BLOCK)
mean_aggregator_kernel(const int* __restrict__ nodes,
                       const int* __restrict__ neigh_ids,
                       const float* __restrict__ neigh_times,
                       const float* __restrict__ features,
                       float* __restrict__ out,
                       int B) {
  // Per-wave private staging: K rows x D floats (16KB per wave, 64KB/block).
  __shared__ __align__(16) float stage[WAVES_PER_BLOCK * K_NEIGH * D_FEAT];

  const int wave = threadIdx.x >> 5;
  const int lane = threadIdx.x & 31;
  const int b    = blockIdx.x * WAVES_PER_BLOCK + wave;
  if (b >= B) return;

  // Streamed-once inputs: non-temporal loads (don't occupy L2).
  const int   base = b * K_NEIGH;
  const int   nid  = __builtin_nontemporal_load(neigh_ids + base + lane);
  const float t    = __builtin_nontemporal_load(neigh_times + base + lane);
  const int   self = __builtin_nontemporal_load(nodes + b);

  // ---- issue all 32 async row gathers first (512B each, contiguous) ------
  // readlane(nid,k) is uniform -> scalar base + lane*16 voffset addressing.
  float* wave_lds = stage + wave * (K_NEIGH * D_FEAT);
#pragma unroll
  for (int k = 0; k < K_NEIGH; ++k) {
    const int    idk  = broadcast_i32(nid, k);
    const float* gsrc = features + (size_t)idk * D_FEAT + lane * 4;
    float*       ldst = wave_lds + k * D_FEAT + lane * 4;
    async_gather_b128(gsrc, ldst);
  }

  // ---- weight math runs in the shadow of the DMA -------------------------
  float w = (t <= T_NOW) ? __expf((t - T_NOW) * INV_TAU) : 0.0f;

  float tot = w;
#pragma unroll
  for (int off = 16; off > 0; off >>= 1) tot += __shfl_xor(tot, off, 32);
  tot = (tot == 0.0f) ? 1.0f : tot;
  const float wn = w / tot;

  // Self feature row overlaps the DMA flight; regular-temporal (table reuse).
  const v4f fself =
      *reinterpret_cast<const v4f*>(features + (size_t)self * D_FEAT + lane * 4);

  // ---- staged consume (async loads retire in order) ----------------------
  v4f   acc  = {0.0f, 0.0f, 0.0f, 0.0f};
  float wsum = 0.0f; // uniform: sum of normalized weights for denom

  wait_asynccnt<16>(); // rows 0..15 resident
#pragma unroll
  for (int k = 0; k < 16; ++k) {
    const float wk = broadcast_f32(wn, k); // SGPR scalar into v_fmac
    wsum += wk;
    const v4f v = *reinterpret_cast<const v4f*>(wave_lds + k * D_FEAT + lane * 4);
    acc += v * wk;
  }

  wait_asynccnt<0>();  // rows 16..31 resident
#pragma unroll
  for (int k = 16; k < K_NEIGH; ++k) {
    const float wk = broadcast_f32(wn, k);
    wsum += wk;
    const v4f v = *reinterpret_cast<const v4f*>(wave_lds + k * D_FEAT + lane * 4);
    acc += v * wk;
  }

  // ---- combine + non-temporal store (written once, never re-read) --------
  const float invd = 1.0f / (1.0f + wsum);
  const v4f o = (fself + acc) * invd;
  __builtin_nontemporal_store(
      o, reinterpret_cast<v4f*>(out + (size_t)b * D_FEAT + lane * 4));
}

extern "C" void kernel_launch(void* const* d_in, const int* in_sizes, int n_in,
                              void* d_out, int out_size, void* d_ws, size_t ws_size,
                              hipStream_t stream) {
  const int*   nodes       = (const int*)d_in[0];
  const int*   neigh_ids   = (const int*)d_in[1];
  const float* neigh_times = (const float*)d_in[2];
  const float* features    = (const float*)d_in[3];
  float*       out         = (float*)d_out;

  const int B = in_sizes[0]; // 50000 nodes; K=32, D=128 fixed by reference
  const dim3 grid((B + WAVES_PER_BLOCK - 1) / WAVES_PER_BLOCK);
  mean_aggregator_kernel<<<grid, BLOCK, 0, stream>>>(nodes, neigh_ids, neigh_times,
                                                     features, out, B);
}